// SamplingStep_67808943669588
// MI455X (gfx1250) — compile-verified
//
#include <hip/hip_runtime.h>

// ---------------------------------------------------------------------------
// Problem constants (static shapes from the reference)
// ---------------------------------------------------------------------------
#define B_DIM   64
#define L_DIM   16
#define D_DIM   1024
#define KHEADS  8
#define MAXOBS  8192

// d_out flat layout: seq | c_sample | c_ent | c_dist
#define OUT_SEQ_OFF   0
#define OUT_SAMP_OFF  (B_DIM * L_DIM * D_DIM)            // 1048576
#define OUT_ENT_OFF   (OUT_SAMP_OFF + B_DIM)             // 1048640
#define OUT_DIST_OFF  (OUT_ENT_OFF + B_DIM)              // 1048704

typedef __attribute__((ext_vector_type(2))) float v2f;
typedef __attribute__((ext_vector_type(8))) float v8f;

// ---------------------------------------------------------------------------
// CDNA5 WMMA: D(16x16,f32) = A(16x4,f32) * B(4x16,f32) + C
// 8-arg class: (neg_a, A, neg_b, B, c_mod, C, reuse_a, reuse_b)
// ---------------------------------------------------------------------------
__device__ __forceinline__ v8f wmma_f32_16x16x4(v2f a, v2f b, v8f c) {
  return __builtin_amdgcn_wmma_f32_16x16x4_f32(
      /*neg_a=*/false, a, /*neg_b=*/false, b,
      /*c_mod=*/(short)0, c, /*reuse_a=*/false, /*reuse_b=*/false);
}

// Faithful port of _obs_key_index select chain (BOUNDARIES = 0..7)
__device__ __forceinline__ int obs_key_index(int index) {
  int ti = index - 1;  // N_TOK_PER_OBS = 1
  int oki = 0;
#pragma unroll
  for (int i = 0; i < KHEADS; ++i) {
    int bnd = KHEADS - 1 - i;
    if (ti < bnd) oki = bnd;
  }
  return oki;
}

// OBS_DIMS = {2048,4096,8192,1024,2048,4096,8192,1024} == 1024 << ((oki+1)&3)
__device__ __forceinline__ int obs_dim(int oki) { return 1024 << ((oki + 1) & 3); }

// ---------------------------------------------------------------------------
// JAX threefry2x32 (exact: 5x4 ARX rounds, key schedule with 0x1BD11BDA)
// ---------------------------------------------------------------------------
__device__ __forceinline__ unsigned rotl32(unsigned v, int d) {
  return (v << d) | (v >> (32 - d));
}
__device__ __forceinline__ void threefry2x32(unsigned ks0, unsigned ks1,
                                             unsigned x0, unsigned x1,
                                             unsigned& o0, unsigned& o1) {
  unsigned ks2 = ks0 ^ ks1 ^ 0x1BD11BDAu;
  x0 += ks0; x1 += ks1;
#define TF_R(r) { x0 += x1; x1 = rotl32(x1, (r)); x1 ^= x0; }
  TF_R(13) TF_R(15) TF_R(26) TF_R(6)   x0 += ks1; x1 += ks2 + 1u;
  TF_R(17) TF_R(29) TF_R(16) TF_R(24)  x0 += ks2; x1 += ks0 + 2u;
  TF_R(13) TF_R(15) TF_R(26) TF_R(6)   x0 += ks0; x1 += ks1 + 3u;
  TF_R(17) TF_R(29) TF_R(16) TF_R(24)  x0 += ks1; x1 += ks2 + 4u;
  TF_R(13) TF_R(15) TF_R(26) TF_R(6)   x0 += ks2; x1 += ks0 + 5u;
#undef TF_R
  o0 = x0; o1 = x1;
}

// ---------------------------------------------------------------------------
// K1: token[64][1024] = seq[:, index, :] @ Wt + bt
// One wave per 16-wide N strip, 4 M-tiles (M=64) accumulated per wave so
// every Wt element is loaded exactly once.
// ---------------------------------------------------------------------------
__global__ void k_token_gemm(const float* __restrict__ seq,
                             const float* __restrict__ Wt,
                             const float* __restrict__ bt,
                             const int* __restrict__ idx_p,
                             float* __restrict__ token) {
  const int idx   = idx_p[0];
  const int lane  = threadIdx.x & 31;
  const int strip = blockIdx.x * (blockDim.x >> 5) + (threadIdx.x >> 5);
  const int n0    = strip << 4;
  if (n0 >= D_DIM) return;
  const int nl = lane & 15;
  const int kb = (lane >> 4) << 1;

  const float bv = bt[n0 + nl];
  v8f acc[4];
#pragma unroll
  for (int t = 0; t < 4; ++t)
#pragma unroll
    for (int r = 0; r < 8; ++r) acc[t][r] = bv;

  const float* Bp = Wt + n0 + nl;
  for (int kk = 0; kk < D_DIM; kk += 4) {
    const int k0 = kk + kb;
    v2f bvec;
    bvec.x = Bp[(size_t)k0 * D_DIM];
    bvec.y = Bp[(size_t)(k0 + 1) * D_DIM];
    if (kk + 36 < D_DIM)
      __builtin_prefetch(Bp + (size_t)(k0 + 32) * D_DIM, 0, 1);
#pragma unroll
    for (int t = 0; t < 4; ++t) {
      const int m = (t << 4) + nl;  // A: M = lane&15 (+ tile), K = kb..kb+1
      v2f avec = *(const v2f*)(seq + ((size_t)m * L_DIM + idx) * D_DIM + k0);
      acc[t] = wmma_f32_16x16x4(avec, bvec, acc[t]);
    }
  }
#pragma unroll
  for (int t = 0; t < 4; ++t)
#pragma unroll
    for (int r = 0; r < 8; ++r) {
      const int m = (t << 4) + ((lane >> 4) << 3) + r;  // D rows r / r+8
      token[(size_t)m * D_DIM + n0 + nl] = acc[t][r];
    }
}

// ---------------------------------------------------------------------------
// K2: c_dist[64][MAXOBS]: cols < od get token @ Wd[oki][:, :od] + bd[oki],
// cols >= od get -inf (the jnp.full padding).  Wd row stride is MAXOBS.
// ---------------------------------------------------------------------------
__global__ void k_dist_gemm(const float* __restrict__ token,
                            const float* __restrict__ Wd,
                            const float* __restrict__ bd,
                            const int* __restrict__ idx_p,
                            float* __restrict__ cdist) {
  const int oki   = obs_key_index(idx_p[0]);
  const int od    = obs_dim(oki);
  const int lane  = threadIdx.x & 31;
  const int strip = blockIdx.x * (blockDim.x >> 5) + (threadIdx.x >> 5);
  const int n0    = strip << 4;
  if (n0 >= MAXOBS) return;
  const int nl = lane & 15;

  if (n0 >= od) {  // -inf padding region
    const float ninf = -__builtin_inff();
#pragma unroll 4
    for (int m = (lane >> 4); m < B_DIM; m += 2)
      cdist[(size_t)m * MAXOBS + n0 + nl] = ninf;
    return;
  }

  const int kb = (lane >> 4) << 1;
  const float bv = bd[oki * MAXOBS + n0 + nl];
  v8f acc[4];
#pragma unroll
  for (int t = 0; t < 4; ++t)
#pragma unroll
    for (int r = 0; r < 8; ++r) acc[t][r] = bv;

  const float* Bp = Wd + (size_t)oki * D_DIM * MAXOBS + n0 + nl;
  for (int kk = 0; kk < D_DIM; kk += 4) {
    const int k0 = kk + kb;
    v2f bvec;
    bvec.x = Bp[(size_t)k0 * MAXOBS];
    bvec.y = Bp[(size_t)(k0 + 1) * MAXOBS];
    if (kk + 36 < D_DIM)
      __builtin_prefetch(Bp + (size_t)(k0 + 32) * MAXOBS, 0, 1);
#pragma unroll
    for (int t = 0; t < 4; ++t) {
      v2f avec = *(const v2f*)(token + (size_t)((t << 4) + nl) * D_DIM + k0);
      acc[t] = wmma_f32_16x16x4(avec, bvec, acc[t]);
    }
  }
#pragma unroll
  for (int t = 0; t < 4; ++t)
#pragma unroll
    for (int r = 0; r < 8; ++r) {
      const int m = (t << 4) + ((lane >> 4) << 3) + r;
      cdist[(size_t)m * MAXOBS + n0 + nl] = acc[t][r];
    }
}

// ---------------------------------------------------------------------------
// K3: per batch row b: Gumbel-argmax sample (exact JAX threefry chain) +
// softmax entropy.  One 256-thread block per row, LDS tree reductions.
// ---------------------------------------------------------------------------
__global__ void k_sample(const float* __restrict__ cdist,
                         const int* __restrict__ idx_p,
                         const int* __restrict__ seed_p,
                         float* __restrict__ out_samp_f,
                         float* __restrict__ out_ent,
                         int* __restrict__ ws_samp) {
  __shared__ float s_v[256];
  __shared__ int   s_i[256];
  __shared__ float s_m[256];

  const int b   = blockIdx.x;
  const int tid = threadIdx.x;
  const int oki = obs_key_index(idx_p[0]);
  const int od  = obs_dim(oki);

  // rng chain: key=(0,seed); iterations 0..oki each do split(); the oki-th
  // split's second subkey (_rng) feeds the categorical.
  unsigned k0 = 0u, k1 = (unsigned)seed_p[0];
  unsigned rk0 = 0u, rk1 = 0u;
  for (int i = 0; i <= oki; ++i) {
    unsigned a0, a1, c0, c1;
    threefry2x32(k0, k1, 0u, 2u, a0, a1);  // counts [0,1,2,3] split in halves
    threefry2x32(k0, k1, 1u, 3u, c0, c1);
    rk0 = a1; rk1 = c1;  // _rng = (y1(0,2), y1(1,3))
    k0  = a0; k1  = c0;  //  rng = (y0(0,2), y0(1,3))
  }

  // Gumbel bits: counter-mode over n = 64*od words; half = 32*od.
  const unsigned half = (unsigned)(B_DIM / 2) * (unsigned)od;
  const unsigned base = (unsigned)b * (unsigned)od;
  const bool lo_half = (b < (B_DIM / 2));
  const float TINY = 1.17549435e-38f;

  float bestv = -__builtin_inff();
  int   besti = 0;
  float maxd  = -__builtin_inff();
  const float* row = cdist + (size_t)b * MAXOBS;

  for (int j = tid; j < od; j += blockDim.x) {
    float d = row[j];
    unsigned flat = base + (unsigned)j;
    unsigned y0, y1, bits;
    if (lo_half) { threefry2x32(rk0, rk1, flat, flat + half, y0, y1); bits = y0; }
    else         { threefry2x32(rk0, rk1, flat - half, flat, y0, y1); bits = y1; }
    float u = __uint_as_float((bits >> 9) | 0x3F800000u) - 1.0f;  // [0,1)
    u = u * (1.0f - TINY) + TINY;
    u = fmaxf(TINY, u);
    float g = -logf(-logf(u));        // Gumbel(0,1), JAX-exact recipe
    float keyv = d + g;               // TEMPERATURE = 1
    if (keyv > bestv) { bestv = keyv; besti = j; }  // first-max within thread
    maxd = fmaxf(maxd, d);
  }
  s_v[tid] = bestv; s_i[tid] = besti; s_m[tid] = maxd;
  __syncthreads();
#pragma unroll
  for (int s = 128; s > 0; s >>= 1) {
    if (tid < s) {
      float v2 = s_v[tid + s]; int i2 = s_i[tid + s];
      if (v2 > s_v[tid] || (v2 == s_v[tid] && i2 < s_i[tid])) {
        s_v[tid] = v2; s_i[tid] = i2;
      }
      s_m[tid] = fmaxf(s_m[tid], s_m[tid + s]);
    }
    __syncthreads();
  }
  const int   samp = s_i[0];
  const float M    = s_m[0];
  __syncthreads();

  // entropy = lse - E[d];  lse = M + log(sum exp(d-M))
  float s1 = 0.0f, s2 = 0.0f;
  for (int j = tid; j < od; j += blockDim.x) {
    float d = row[j];
    float e = expf(d - M);
    s1 += e; s2 += d * e;
  }
  s_v[tid] = s1; s_m[tid] = s2;
  __syncthreads();
#pragma unroll
  for (int s = 128; s > 0; s >>= 1) {
    if (tid < s) { s_v[tid] += s_v[tid + s]; s_m[tid] += s_m[tid + s]; }
    __syncthreads();
  }
  if (tid == 0) {
    float lse = M + logf(s_v[0]);
    out_ent[b]    = lse - s_m[0] / s_v[0];
    out_samp_f[b] = (float)samp;
    ws_samp[b]    = samp;
  }
}

// ---------------------------------------------------------------------------
// K4: out_seq = seq, with row (index+1) replaced by Wu[oki][sample[b]] + bu[oki].
// float4 streaming copy; the gather row is fused in.
// ---------------------------------------------------------------------------
__global__ void k_update_seq(const float* __restrict__ seq,
                             const float* __restrict__ Wu,
                             const float* __restrict__ bu,
                             const int* __restrict__ idx_p,
                             const int* __restrict__ ws_samp,
                             float* __restrict__ out_seq) {
  const int idx = idx_p[0];
  const int oki = obs_key_index(idx);
  size_t e = (size_t)blockIdx.x * blockDim.x + threadIdx.x;  // float4 id
  const size_t total4 = (size_t)B_DIM * L_DIM * D_DIM / 4;
  if (e >= total4) return;
  size_t f = e * 4;
  int d = (int)(f & (D_DIM - 1));
  int l = (int)((f >> 10) & (L_DIM - 1));
  int b = (int)(f >> 14);

  float4 v;
  if (l == idx + 1) {  // (if idx+1 >= L this never triggers: JAX OOB-drop)
    int s = ws_samp[b];
    const float* wr = Wu + ((size_t)oki * MAXOBS + (size_t)s) * D_DIM + d;
    const float* br = bu + (size_t)oki * D_DIM + d;
    v = make_float4(wr[0] + br[0], wr[1] + br[1], wr[2] + br[2], wr[3] + br[3]);
  } else {
    v = *(const float4*)(seq + f);
  }
  *(float4*)(out_seq + f) = v;
}

// ---------------------------------------------------------------------------
// Host launcher
// ---------------------------------------------------------------------------
extern "C" void kernel_launch(void* const* d_in, const int* in_sizes, int n_in,
                              void* d_out, int out_size, void* d_ws, size_t ws_size,
                              hipStream_t stream) {
  const float* seq  = (const float*)d_in[0];
  const float* Wt   = (const float*)d_in[1];
  const float* bt   = (const float*)d_in[2];
  const float* Wd   = (const float*)d_in[3];
  const float* bd   = (const float*)d_in[4];
  const float* Wu   = (const float*)d_in[5];
  const float* bu   = (const float*)d_in[6];
  const int*   idx  = (const int*)d_in[7];
  const int*   seed = (const int*)d_in[8];

  float* out      = (float*)d_out;
  float* out_seq  = out + OUT_SEQ_OFF;
  float* out_samp = out + OUT_SAMP_OFF;
  float* out_ent  = out + OUT_ENT_OFF;
  float* out_dist = out + OUT_DIST_OFF;

  float* ws_token = (float*)d_ws;                                   // 256 KB
  int*   ws_samp  = (int*)((char*)d_ws + (size_t)B_DIM * D_DIM * 4); // 256 B

  // K1: 64 N-strips, 8 waves/block -> 8 blocks
  k_token_gemm<<<8, 256, 0, stream>>>(seq, Wt, bt, idx, ws_token);
  // K2: 512 N-strips over MAXOBS -> 64 blocks
  k_dist_gemm<<<64, 256, 0, stream>>>(ws_token, Wd, bd, idx, out_dist);
  // K3: one block per batch row
  k_sample<<<B_DIM, 256, 0, stream>>>(out_dist, idx, seed, out_samp, out_ent, ws_samp);
  // K4: 64*16*1024/4 float4s / 256 threads -> 1024 blocks
  k_update_seq<<<1024, 256, 0, stream>>>(seq, Wu, bu, idx, ws_samp, out_seq);
}